// NeuralSemiLagrangian_39144331936088
// MI455X (gfx1250) — compile-verified
//
#include <hip/hip_runtime.h>

// ---------------- problem constants (from reference setup) ----------------
constexpr int BB   = 2;
constexpr int HID  = 128;
constexpr int NVC  = 16;
constexpr int HH   = 361;
constexpr int WW   = 720;
constexpr int HWSZ = HH * WW;            // 259920
constexpr int PADQ = 2;
constexpr int HP   = HH + 2 * PADQ;      // 365
constexpr int WP   = WW + 2 * PADQ;      // 724
constexpr int NTILES = HWSZ / 16;        // 16245 (exact)
constexpr float ACUB = -0.75f;
constexpr float TWO_PI_F = 6.28318530717958647692f;

typedef __attribute__((ext_vector_type(2))) float v2f;
typedef __attribute__((ext_vector_type(8))) float v8f;

// ---------------- down projection: 128 -> 16 via WMMA f32 16x16x4 ----------
// One wave per 16-position tile. M = 16 output channels, N = 16 pixels,
// K = 128 hidden channels stepped by 4.
__global__ __launch_bounds__(128) void k_down(const float* __restrict__ hf,
                                              const float* __restrict__ dwn_w,
                                              const float* __restrict__ dwn_b,
                                              float* __restrict__ proj) {
  const int lane = threadIdx.x & 31;
  const int wv   = threadIdx.x >> 5;
  const int tile = blockIdx.x * 4 + wv;
  const int b    = blockIdx.y;
  if (tile >= NTILES) return;                 // wave-uniform: EXEC stays all-1
  const int n  = lane & 15;
  const int hi = lane >> 4;                   // 0: K,K+1   1: K+2,K+3
  const size_t pos0 = (size_t)tile * 16;
  const float* hfb = hf + (size_t)b * HID * HWSZ + pos0 + n;
  v8f acc = {0.f, 0.f, 0.f, 0.f, 0.f, 0.f, 0.f, 0.f};
#pragma unroll
  for (int k = 0; k < HID; k += 4) {
    const int ka = k + (hi << 1);
    v2f A, Bm;
    A.x  = dwn_w[n * HID + ka];               // A[M=n, K=ka]
    A.y  = dwn_w[n * HID + ka + 1];
    Bm.x = hfb[(size_t)ka * HWSZ];            // B[K=ka, N=n]
    Bm.y = hfb[(size_t)(ka + 1) * HWSZ];
    acc = __builtin_amdgcn_wmma_f32_16x16x4_f32(false, A, false, Bm,
                                                (short)0, acc, false, false);
  }
#pragma unroll
  for (int r = 0; r < 8; ++r) {
    const int ch = r + (hi << 3);             // D vgpr r: M=r (lo) / r+8 (hi)
    proj[((size_t)b * NVC + ch) * HWSZ + pos0 + n] = acc[r] + dwn_b[ch];
  }
}

// ---------------- pole fix: replace row 0 and row H-1 by longitude mean ----
// grid = (#b * #c * 2) blocks; channel count is implicit in block count.
__global__ __launch_bounds__(256) void k_polefix(float* __restrict__ x) {
  const int id   = blockIdx.x;
  const int pole = id & 1;
  const int bc   = id >> 1;                   // b*C + c
  const int row  = pole ? (HH - 1) : 0;
  float* p = x + (size_t)bc * HWSZ + (size_t)row * WW;
  __shared__ float sm[256];
  float s = 0.f;
  for (int j = threadIdx.x; j < WW; j += 256) s += p[j];
  sm[threadIdx.x] = s;
  __syncthreads();
  for (int off = 128; off > 0; off >>= 1) {
    if (threadIdx.x < off) sm[threadIdx.x] += sm[threadIdx.x + off];
    __syncthreads();
  }
  const float mean = sm[0] * (1.0f / (float)WW);
  for (int j = threadIdx.x; j < WW; j += 256) p[j] = mean;
}

// ---------------- geo-cyclic pad of proj -> padded [B*NV, HP, WP] ----------
__global__ __launch_bounds__(256) void k_geopad(const float* __restrict__ proj,
                                                float* __restrict__ pad) {
  const long idx = (long)blockIdx.x * 256 + threadIdx.x;
  const long total = (long)BB * NVC * HP * WP;
  if (idx >= total) return;
  const int xp  = (int)(idx % WP);
  const long t  = idx / WP;
  const int yp  = (int)(t % HP);
  const int nim = (int)(t / HP);              // b*NV + v
  int xs = xp - PADQ;
  if (xs < 0) xs += WW; else if (xs >= WW) xs -= WW;
  int ys;
  if (yp < PADQ) {                            // top: flip rows + half-world roll
    ys = PADQ - 1 - yp;
    xs += WW / 2; if (xs >= WW) xs -= WW;
  } else if (yp >= PADQ + HH) {               // bottom
    ys = 2 * HH - 1 + PADQ - yp;
    xs += WW / 2; if (xs >= WW) xs -= WW;
  } else {
    ys = yp - PADQ;
  }
  pad[idx] = proj[((size_t)nim * HH + ys) * WW + xs];
}

// ---------------- cubic convolution weights (a = -0.75) --------------------
__device__ __forceinline__ void cubic_w(float tt, float w[4]) {
  const float t1 = tt + 1.0f;
  const float t2 = 1.0f - tt;
  const float t3 = 2.0f - tt;
  w[0] = ((ACUB * t1 - 5.0f * ACUB) * t1 + 8.0f * ACUB) * t1 - 4.0f * ACUB;
  w[1] = ((ACUB + 2.0f) * tt - (ACUB + 3.0f)) * tt * tt + 1.0f;
  w[2] = ((ACUB + 2.0f) * t2 - (ACUB + 3.0f)) * t2 * t2 + 1.0f;
  w[3] = ((ACUB * t3 - 5.0f * ACUB) * t3 + 8.0f * ACUB) * t3 - 4.0f * ACUB;
}

// ---------------- departure point + bicubic gather + depthwise -------------
__global__ __launch_bounds__(256) void k_advect(const float* __restrict__ u,
                                                const float* __restrict__ v,
                                                const float* __restrict__ dtp,
                                                const float* __restrict__ latg,
                                                const float* __restrict__ lngg,
                                                const float* __restrict__ dww,
                                                const float* __restrict__ dwb,
                                                const float* __restrict__ pad,
                                                float* __restrict__ yout) {
  const long idx = (long)blockIdx.x * 256 + threadIdx.x;
  const long total = (long)BB * NVC * HWSZ;
  if (idx >= total) return;
  const int w   = (int)(idx % WW);
  const long t  = idx / WW;
  const int h   = (int)(t % HH);
  const int nim = (int)(t / HH);              // b*NV + v
  const int vch = nim & (NVC - 1);

  const float dt = dtp[0];
  const float lon_pr = -u[idx] * dt;
  const float lat_pr = -v[idx] * dt;
  const int gi = h * WW + w;
  const float lat_g = latg[gi];
  const float lon_g = lngg[gi];

  float s_lat_pr, c_lat_pr, s_lon_pr, c_lon_pr, s_lat_g, c_lat_g;
  sincosf(lat_pr, &s_lat_pr, &c_lat_pr);
  sincosf(lon_pr, &s_lon_pr, &c_lon_pr);
  sincosf(lat_g, &s_lat_g, &c_lat_g);

  float sin_lat = s_lat_pr * c_lat_g + c_lat_pr * c_lon_pr * s_lat_g;
  sin_lat = fminf(fmaxf(sin_lat, -1.0f + 1e-7f), 1.0f - 1e-7f);
  const float lat_dep = asinf(sin_lat);
  const float num = c_lat_pr * s_lon_pr;
  const float den = c_lat_pr * c_lon_pr * c_lat_g - s_lat_pr * s_lat_g;
  float lon_dep = lon_g + atan2f(num, den);
  lon_dep = fmodf(lon_dep + TWO_PI_F, TWO_PI_F);
  if (lon_dep < 0.f) lon_dep += TWO_PI_F;

  const float min_lat = latg[0];
  const float max_lat = latg[(HH - 1) * WW];
  const float min_lon = lngg[0];
  const float max_lon = lngg[WW - 1];
  const float pix_x = (lon_dep - min_lon) / (max_lon - min_lon) * (float)(WW - 1);
  const float pix_y = (lat_dep - min_lat) / (max_lat - min_lat) * (float)(HH - 1);
  // normalize/unnormalize round trip exactly as reference (align_corners=True)
  const float gx = 2.0f * ((pix_x + (float)PADQ) / (float)(WP - 1)) - 1.0f;
  const float gy = 2.0f * ((pix_y + (float)PADQ) / (float)(HP - 1)) - 1.0f;
  const float ix = (gx + 1.0f) * 0.5f * (float)(WP - 1);
  const float iy = (gy + 1.0f) * 0.5f * (float)(HP - 1);

  const float x0f = floorf(ix), y0f = floorf(iy);
  const float tx = ix - x0f, ty = iy - y0f;
  const int x0 = (int)x0f, y0 = (int)y0f;
  float wx[4], wy[4];
  cubic_w(tx, wx);
  cubic_w(ty, wy);

  const float* img = pad + (size_t)nim * HP * WP;
  float acc = 0.f;
#pragma unroll
  for (int dy = 0; dy < 4; ++dy) {
    const int yy = y0 + dy - 1;
    const bool vy = (yy >= 0) && (yy < HP);
    const int yyc = yy < 0 ? 0 : (yy > HP - 1 ? HP - 1 : yy);
    const float* row = img + (size_t)yyc * WP;
#pragma unroll
    for (int dx = 0; dx < 4; ++dx) {
      const int xx = x0 + dx - 1;
      const bool vx = (xx >= 0) && (xx < WP);
      const int xxc = xx < 0 ? 0 : (xx > WP - 1 ? WP - 1 : xx);
      acc += row[xxc] * (wy[dy] * wx[dx] * ((vy && vx) ? 1.0f : 0.0f));
    }
  }
  yout[idx] = acc * dww[vch] + dwb[vch];
}

// ---------------- up projection: 16 -> 128 via WMMA f32 16x16x4 ------------
// 8 waves per block, one 16-output-channel tile each; N = 16 pixels, K = 16.
__global__ __launch_bounds__(256) void k_up(const float* __restrict__ yv,
                                            const float* __restrict__ up_w,
                                            const float* __restrict__ up_b,
                                            float* __restrict__ out) {
  const int lane  = threadIdx.x & 31;
  const int ctile = threadIdx.x >> 5;         // 0..7
  const int tile  = blockIdx.x;
  const int b     = blockIdx.y;
  const int n  = lane & 15;
  const int hi = lane >> 4;
  const size_t pos0 = (size_t)tile * 16;
  const float* yb = yv + (size_t)b * NVC * HWSZ + pos0 + n;
  v8f acc = {0.f, 0.f, 0.f, 0.f, 0.f, 0.f, 0.f, 0.f};
#pragma unroll
  for (int k = 0; k < NVC; k += 4) {
    const int ka = k + (hi << 1);
    v2f A, Bm;
    A.x  = up_w[(ctile * 16 + n) * NVC + ka]; // A[M, K] = up_w[o, v]
    A.y  = up_w[(ctile * 16 + n) * NVC + ka + 1];
    Bm.x = yb[(size_t)ka * HWSZ];             // B[K, N] = y[b, v, pos]
    Bm.y = yb[(size_t)(ka + 1) * HWSZ];
    acc = __builtin_amdgcn_wmma_f32_16x16x4_f32(false, A, false, Bm,
                                                (short)0, acc, false, false);
  }
#pragma unroll
  for (int r = 0; r < 8; ++r) {
    const int ch = ctile * 16 + r + (hi << 3);
    out[((size_t)b * HID + ch) * HWSZ + pos0 + n] = acc[r] + up_b[ch];
  }
}

// ---------------------------------------------------------------------------
extern "C" void kernel_launch(void* const* d_in, const int* in_sizes, int n_in,
                              void* d_out, int out_size, void* d_ws, size_t ws_size,
                              hipStream_t stream) {
  const float* hf     = (const float*)d_in[0];
  const float* u      = (const float*)d_in[1];
  const float* v      = (const float*)d_in[2];
  const float* dt     = (const float*)d_in[3];
  const float* latg   = (const float*)d_in[4];
  const float* lngg   = (const float*)d_in[5];
  const float* down_w = (const float*)d_in[6];
  const float* down_b = (const float*)d_in[7];
  const float* dw_w   = (const float*)d_in[8];
  const float* dw_b   = (const float*)d_in[9];
  const float* up_w   = (const float*)d_in[10];
  const float* up_b   = (const float*)d_in[11];
  float* out = (float*)d_out;

  // workspace layout (floats): [padded : B*NV*HP*WP][proj / y : B*NV*H*W]
  float* padbuf = (float*)d_ws;
  float* proj   = padbuf + (size_t)BB * NVC * HP * WP;   // reused as y after pad

  dim3 gdown((NTILES + 3) / 4, BB);
  k_down<<<gdown, 128, 0, stream>>>(hf, down_w, down_b, proj);

  k_polefix<<<BB * NVC * 2, 256, 0, stream>>>(proj);

  const long padtot = (long)BB * NVC * HP * WP;
  k_geopad<<<(int)((padtot + 255) / 256), 256, 0, stream>>>(proj, padbuf);

  const long atot = (long)BB * NVC * HWSZ;
  k_advect<<<(int)((atot + 255) / 256), 256, 0, stream>>>(
      u, v, dt, latg, lngg, dw_w, dw_b, padbuf, proj);

  dim3 gup(NTILES, BB);
  k_up<<<gup, 256, 0, stream>>>(proj, up_w, up_b, out);

  k_polefix<<<BB * HID * 2, 256, 0, stream>>>(out);
}